// MHFFNMoEMultiHeadFFNFlash_59219009077890
// MI455X (gfx1250) — compile-verified
//
#include <hip/hip_runtime.h>
#include <math.h>

// ---------------------------------------------------------------------------
// MI455X / gfx1250 implementation of MHFFNMoEMultiHeadFFNFlash forward.
// All GEMMs use V_WMMA_F32_16X16X4_F32 (full f32 precision, matrix pipe).
// M=32 blocking: every B fragment feeds two WMMAs (two 16-row accumulators),
// halving cache-side operand traffic per FLOP vs one-tile-per-wave.
// ---------------------------------------------------------------------------

typedef __attribute__((ext_vector_type(2))) float v2f;
typedef __attribute__((ext_vector_type(4))) float v4f;
typedef __attribute__((ext_vector_type(8))) float v8f;

#define WMMA_F32(a, b, c) \
  __builtin_amdgcn_wmma_f32_16x16x4_f32(false, (a), false, (b), (short)0, (c), false, false)

// Problem constants (compile-time; matches reference shapes)
constexpr int S    = 1024;
constexpr int HID  = 2048;
constexpr int H    = 16;
constexpr int D    = 128;
constexpr int E    = 32;
constexpr int KUVW = 512;   // KUV * D (row width of gate/up/down weights)
constexpr float EPS           = 1e-6f;
constexpr float HEAD_SCALING  = 4.0f;                    // sqrt(H)
constexpr float FINAL_SCALING = 0.6123724356957945f;     // sqrt(3/8)

// LDS row stride: 132 floats. 132 mod 64 == 4, so the WMMA-A read pattern
// (lane l reads floats m*132 + k + (l>>4)*2 + {0,1}, m = l&15) maps the 64
// floats of a wave's request onto all 64 LDS banks exactly once.
constexpr int LSTR = 132;

// ---------------------------------------------------------------------------
// C[M,N] = A[M,K] @ B[N,K]^T   (nn.Linear: out = x @ W^T)
// grid: (N/128, M/32); block: 256 threads = 8 waves; wave w -> n-tile w.
// Each wave owns two 16x16 output tiles (rows mbase..+15 and mbase+16..+31),
// so every global B fragment is consumed by two WMMAs.
// ---------------------------------------------------------------------------
__global__ __launch_bounds__(256) void gemm_nt_wmma(const float* __restrict__ A,
                                                    const float* __restrict__ Bw,
                                                    float* __restrict__ C,
                                                    int M, int N, int K) {
  __shared__ float As[32 * LSTR];
  const int tid   = threadIdx.x;
  const int lane  = tid & 31;
  const int wv    = tid >> 5;
  const int mbase = blockIdx.y * 32;
  const int nb    = blockIdx.x * 128 + wv * 16;
  const int mrow  = lane & 15;
  const int khalf = (lane >> 4) * 2;

  const float* Brow = Bw + (size_t)(nb + mrow) * K;  // lane's B row (n fixed)
  v8f acc0 = {}, acc1 = {};

  for (int kc = 0; kc < K; kc += 128) {
    __syncthreads();
    // cooperative, coalesced A-tile stage: 32 rows x 128 k
    for (int i = tid; i < 32 * 128; i += 256) {
      int r = i >> 7, c = i & 127;
      As[r * LSTR + c] = A[(size_t)(mbase + r) * K + kc + c];
    }
    __syncthreads();
    const float* bp = Brow + kc + khalf;
    if (kc + 128 < K) __builtin_prefetch(Brow + kc + 128, 0, 1);  // global_prefetch_b8
#pragma unroll
    for (int k = 0; k < 128; k += 4) {
      v2f b  = *(const v2f*)(bp + k);                             // B[k+khalf..][n]
      v2f a0 = *(const v2f*)&As[mrow * LSTR + k + khalf];         // rows 0..15
      v2f a1 = *(const v2f*)&As[(16 + mrow) * LSTR + k + khalf];  // rows 16..31
      acc0 = WMMA_F32(a0, b, acc0);
      acc1 = WMMA_F32(a1, b, acc1);
    }
  }
  // C/D layout: VGPR v -> row m = v + 8*(lane>>4), col n = lane&15
  const int m0  = (lane >> 4) << 3;
  const int col = nb + mrow;
#pragma unroll
  for (int v = 0; v < 8; ++v) {
    C[(size_t)(mbase + m0 + v) * N + col]      = acc0[v];
    C[(size_t)(mbase + 16 + m0 + v) * N + col] = acc1[v];
  }
}

// ---------------------------------------------------------------------------
// Per-head RMSNorm: Q[s, h*128+d] -> Qn[h][s][d]  (fp32, eps inside rsqrt)
// One wave per (s,h) row; 4 elements per lane; shuffle reduction.
// ---------------------------------------------------------------------------
__global__ __launch_bounds__(256) void rmsnorm_k(const float* __restrict__ Q,
                                                 const float* __restrict__ qnw,
                                                 float* __restrict__ Qn) {
  const int gw   = blockIdx.x * 8 + (threadIdx.x >> 5);
  const int lane = threadIdx.x & 31;
  const int s = gw >> 4;
  const int h = gw & 15;
  const float* src = Q + (size_t)s * HID + h * D + lane * 4;
  v4f q = *(const v4f*)src;
  float ss = q[0] * q[0] + q[1] * q[1] + q[2] * q[2] + q[3] * q[3];
#pragma unroll
  for (int off = 16; off; off >>= 1) ss += __shfl_xor(ss, off, 32);
  const float rs = rsqrtf(ss * (1.0f / 128.0f) + EPS);
  v4f w = *(const v4f*)(qnw + h * D + lane * 4);
  v4f o;
  o[0] = q[0] * rs * w[0];
  o[1] = q[1] * rs * w[1];
  o[2] = q[2] * rs * w[2];
  o[3] = q[3] * rs * w[3];
  *(v4f*)(Qn + ((size_t)h * S + s) * D + lane * 4) = o;
}

// ---------------------------------------------------------------------------
// Router: R[h][s][e] = sigmoid(Qn[h][s][:] . router[h][:][e]) normalized
// over experts. One wave per (h,s); lane == expert (E==32==wave32).
// ---------------------------------------------------------------------------
__global__ __launch_bounds__(256) void router_k(const float* __restrict__ Qn,
                                                const float* __restrict__ router,
                                                float* __restrict__ R) {
  const int gw   = blockIdx.x * 8 + (threadIdx.x >> 5);
  const int lane = threadIdx.x & 31;  // expert index
  const int h = gw >> 10;
  const int s = gw & 1023;
  const float* q  = Qn + ((size_t)h * S + s) * D;
  const float* rt = router + (size_t)h * D * E + lane;
  float acc = 0.0f;
#pragma unroll 4
  for (int d = 0; d < D; ++d) acc = fmaf(q[d], rt[(size_t)d * E], acc);
  float r = 1.0f / (1.0f + __expf(-acc));  // sigmoid
  float tot = r;
#pragma unroll
  for (int off = 16; off; off >>= 1) tot += __shfl_xor(tot, off, 32);
  R[((size_t)h * S + s) * E + lane] = r / fmaxf(tot, 1e-20f);
}

// ---------------------------------------------------------------------------
// Fused MoE FFN per head. grid: (S/32, H); block: 256 = 8 waves.
// 32 s-rows per block (two 16-row WMMA tiles per wave). Per expert chunk
// (128 rows of I): gate/up WMMA -> silu*up*R -> LDS hidden -> down-projection
// WMMA accumulation. Output never leaves the 16 accumulator VGPRs until the
// final scaled store; hidden (256 MB if materialized) never touches global.
// ---------------------------------------------------------------------------
__global__ __launch_bounds__(256) void ffn_moe_wmma(const float* __restrict__ Qn,
                                                    const float* __restrict__ Rw,
                                                    const float* __restrict__ gwt,
                                                    const float* __restrict__ uwt,
                                                    const float* __restrict__ dwt,
                                                    float* __restrict__ mid) {
  __shared__ float Qs[32 * LSTR];
  __shared__ float Hs[32 * LSTR];
  __shared__ float Rs[32];

  const int tid   = threadIdx.x;
  const int lane  = tid & 31;
  const int wv    = tid >> 5;
  const int h     = blockIdx.y;
  const int kuv   = h >> 2;            // head -> kuv group (GROUPS = 4)
  const int sbase = blockIdx.x * 32;
  const int mrow  = lane & 15;
  const int khalf = (lane >> 4) * 2;
  const int m0    = (lane >> 4) << 3;  // C/D row base for this half-wave

  // stage Q tile (32 x 128) into LDS, conflict-free stride
  const float* qb = Qn + ((size_t)h * S + sbase) * D;
  for (int i = tid; i < 32 * 128; i += 256) {
    int r = i >> 7, c = i & 127;
    Qs[r * LSTR + c] = qb[r * 128 + c];
  }
  __syncthreads();

  // per-lane weight base pointers
  // gate/up B: B[k][n] = W[chunk*128 + wv*16 + n][kuv*128 + k]
  const float* gp0 = gwt + (size_t)(wv * 16 + mrow) * KUVW + kuv * 128 + khalf;
  const float* up0 = uwt + (size_t)(wv * 16 + mrow) * KUVW + kuv * 128 + khalf;
  // down B: B[k][n] = V[chunk*128 + k][kuv*128 + wv*16 + n]
  const float* dp0 = dwt + (size_t)khalf * KUVW + kuv * 128 + wv * 16 + mrow;

  v8f acc0 = {}, acc1 = {};

  for (int chunk = 0; chunk < E; ++chunk) {
    if (tid < 32)  // router scale for the 32 s-rows of this tile, this expert
      Rs[tid] = Rw[((size_t)h * S + sbase + tid) * E + chunk];

    v8f g0 = {}, g1 = {}, u0 = {}, u1 = {};
    const float* gp = gp0 + (size_t)chunk * 128 * KUVW;
    const float* up = up0 + (size_t)chunk * 128 * KUVW;
#pragma unroll
    for (int k = 0; k < 128; k += 4) {
      v2f bg = *(const v2f*)(gp + k);
      v2f bu = *(const v2f*)(up + k);
      v2f a0 = *(const v2f*)&Qs[mrow * LSTR + k + khalf];
      v2f a1 = *(const v2f*)&Qs[(16 + mrow) * LSTR + k + khalf];
      g0 = WMMA_F32(a0, bg, g0);
      g1 = WMMA_F32(a1, bg, g1);
      u0 = WMMA_F32(a0, bu, u0);
      u1 = WMMA_F32(a1, bu, u1);
    }

    __syncthreads();  // Rs visible; previous chunk's Hs reads complete
#pragma unroll
    for (int v = 0; v < 8; ++v) {
      float gv0 = g0[v] * HEAD_SCALING;
      float uv0 = u0[v] * HEAD_SCALING;
      float gv1 = g1[v] * HEAD_SCALING;
      float uv1 = u1[v] * HEAD_SCALING;
      // silu(gate) * up * router
      float hv0 = (gv0 / (1.0f + __expf(-gv0))) * uv0 * Rs[m0 + v];
      float hv1 = (gv1 / (1.0f + __expf(-gv1))) * uv1 * Rs[16 + m0 + v];
      Hs[(m0 + v) * LSTR + wv * 16 + mrow]      = hv0;
      Hs[(16 + m0 + v) * LSTR + wv * 16 + mrow] = hv1;
    }
    __syncthreads();  // hidden tile complete

    const float* dp = dp0 + (size_t)chunk * 128 * KUVW;
#pragma unroll
    for (int k = 0; k < 128; k += 4) {
      v2f bd;
      bd.x = dp[(size_t)k * KUVW];
      bd.y = dp[(size_t)(k + 1) * KUVW];
      v2f a0 = *(const v2f*)&Hs[mrow * LSTR + k + khalf];
      v2f a1 = *(const v2f*)&Hs[(16 + mrow) * LSTR + k + khalf];
      acc0 = WMMA_F32(a0, bd, acc0);
      acc1 = WMMA_F32(a1, bd, acc1);
    }
  }

  // mid[s][h*128 + col] layout so the final projection is a plain GEMM
  const int col = h * D + wv * 16 + mrow;
#pragma unroll
  for (int v = 0; v < 8; ++v) {
    mid[(size_t)(sbase + m0 + v) * HID + col]      = acc0[v] * FINAL_SCALING;
    mid[(size_t)(sbase + 16 + m0 + v) * HID + col] = acc1[v] * FINAL_SCALING;
  }
}

// ---------------------------------------------------------------------------
extern "C" void kernel_launch(void* const* d_in, const int* in_sizes, int n_in,
                              void* d_out, int out_size, void* d_ws, size_t ws_size,
                              hipStream_t stream) {
  const float* x      = (const float*)d_in[0];
  const float* w1     = (const float*)d_in[1];
  const float* w2     = (const float*)d_in[2];
  const float* qnw    = (const float*)d_in[3];
  const float* gwt    = (const float*)d_in[4];
  const float* uwt    = (const float*)d_in[5];
  const float* dwt    = (const float*)d_in[6];
  const float* router = (const float*)d_in[7];
  float* out = (float*)d_out;

  float* ws  = (float*)d_ws;
  float* Q   = ws;                                  // 1024*2048
  float* Qn  = ws + (size_t)2 * 1024 * 1024;        // 16*1024*128
  float* R   = ws + (size_t)4 * 1024 * 1024;        // 16*1024*32
  float* mid = ws + (size_t)4 * 1024 * 1024 + 512 * 1024;  // 1024*2048

  // 1) Q = x @ w_self^T
  gemm_nt_wmma<<<dim3(HID / 128, S / 32), 256, 0, stream>>>(x, w1, Q, S, HID, HID);
  // 2) per-head RMSNorm, transpose to [h][s][d]
  rmsnorm_k<<<(S * H) / 8, 256, 0, stream>>>(Q, qnw, Qn);
  // 3) router probabilities
  router_k<<<(S * H) / 8, 256, 0, stream>>>(Qn, router, R);
  // 4) fused per-head MoE FFN (32 s-rows per block)
  ffn_moe_wmma<<<dim3(S / 32, H), 256, 0, stream>>>(Qn, R, gwt, uwt, dwt, mid);
  // 5) out = mid @ w_self_2^T
  gemm_nt_wmma<<<dim3(HID / 128, S / 32), 256, 0, stream>>>(mid, w2, out, S, HID, HID);
}